// Net_5239860101629
// MI455X (gfx1250) — compile-verified
//
#include <hip/hip_runtime.h>

#define D 128
#define BN_EPS 1e-5f

typedef __attribute__((ext_vector_type(2))) float v2f;
typedef __attribute__((ext_vector_type(8))) float v8f;
typedef __attribute__((ext_vector_type(4))) unsigned int u32x4;
typedef __attribute__((ext_vector_type(8))) int i32x8;
typedef __attribute__((ext_vector_type(4))) int i32x4;

#if __has_builtin(__builtin_amdgcn_tensor_load_to_lds) && __has_builtin(__builtin_amdgcn_s_wait_tensorcnt)
#define HAVE_TDM 1
#else
#define HAVE_TDM 0
#endif

__device__ __forceinline__ v8f wmma4(v2f a, v2f b, v8f c) {
    // D(16x16,f32) = A(16x4,f32) x B(4x16,f32) + C
    return __builtin_amdgcn_wmma_f32_16x16x4_f32(
        /*neg_a=*/false, a, /*neg_b=*/false, b,
        /*c_mod=*/(short)0, c, /*reuse_a=*/false, /*reuse_b=*/false);
}

// ---- stage the 128x128 fp32 weight into LDS (offset 0) via the TDM --------
// D# per CDNA5 ISA ch.8: group0 = {count|flags, lds_addr, global_addr, type},
// group1 = {mask|data_size, dims/strides/tile dims}. 2-D tensor -> groups 2/3
// disabled (zero). Issued by wave 0 only (TDM ignores EXEC; one DMA per WG).
__device__ __forceinline__ void stage_weights(float* sW, const float* W) {
#if HAVE_TDM
    if ((threadIdx.x >> 5) == 0) {
        unsigned long long ga = (unsigned long long)W;
        unsigned alo = (unsigned)__builtin_amdgcn_readfirstlane((int)(ga & 0xFFFFFFFFull));
        unsigned ahi = (unsigned)__builtin_amdgcn_readfirstlane((int)(ga >> 32));
        u32x4 g0;
        g0[0] = 1u;                                   // count=1 (valid user D#)
        g0[1] = 0u;                                   // lds_addr = 0 (sW is sole LDS obj)
        g0[2] = alo;                                  // global_addr[31:0]
        g0[3] = (ahi & 0x01FFFFFFu) | 0x80000000u;    // global_addr[56:32] | type=2
        i32x8 g1;
        g1[0] = (int)(2u << 16);                      // workgroup_mask=0, data_size=4B
        g1[1] = (int)(128u << 16);                    // barrier_addr=0 | tensor_dim0=128
        g1[2] = (int)(128u << 16);                    // dim0_hi=0 | tensor_dim1=128
        g1[3] = (int)(128u << 16);                    // dim1_hi=0 | tile_dim0=128
        g1[4] = 128;                                  // tile_dim1=128, tile_dim2=0
        g1[5] = 128;                                  // tensor_dim0_stride=128
        g1[6] = 0;                                    // stride hi / dim1_stride lo
        g1[7] = 0;
        i32x4 gz = {0, 0, 0, 0};
#if defined(__clang_major__) && (__clang_major__ >= 23)
        i32x8 gz8 = {0, 0, 0, 0, 0, 0, 0, 0};
        __builtin_amdgcn_tensor_load_to_lds(g0, g1, gz, gz, gz8, 0);
#else
        __builtin_amdgcn_tensor_load_to_lds(g0, g1, gz, gz, 0);
#endif
        __builtin_amdgcn_s_wait_tensorcnt(0);
    }
    // The TDM builtin carries no pointer the optimizer can alias-track, so
    // sW has no visible stores; without this, loads from sW fold to undef.
    // Escape the pointer through an opaque asm that may write memory.
    {
        void* p = (void*)sW;
        asm volatile("" : "+v"(p) : : "memory");
    }
    __syncthreads();
#else
    for (int i = threadIdx.x; i < (D * D) / 4; i += (int)blockDim.x)
        reinterpret_cast<float4*>(sW)[i] = reinterpret_cast<const float4*>(W)[i];
    __syncthreads();
#endif
}

// ---------------------------------------------------------------- zero fill
__global__ void k_zero(float* __restrict__ p, int n) {
    int i = blockIdx.x * blockDim.x + threadIdx.x;
    int stride = gridDim.x * blockDim.x;
    for (; i < n; i += stride) p[i] = 0.f;
}

// ------------------------------------------------- scatter-add aggregation
// one wave per edge: 32 lanes x float4 = 128 features (x + agg stay L2-resident)
__global__ void k_scatter(const float* __restrict__ x, const int* __restrict__ src,
                          const int* __restrict__ dst, float* __restrict__ agg, int nE) {
    int gid = blockIdx.x * blockDim.x + threadIdx.x;
    int e = gid >> 5;
    int lane = gid & 31;
    if (e >= nE) return;
    int s = src[e], d = dst[e];
    const float4 v = reinterpret_cast<const float4*>(x + (size_t)s * D)[lane];
    float* p = agg + (size_t)d * D + lane * 4;
    atomicAdd(p + 0, v.x);
    atomicAdd(p + 1, v.y);
    atomicAdd(p + 2, v.z);
    atomicAdd(p + 3, v.w);
}

// ---------------------- out = l2norm(agg) + (1+eps)*l2norm(x), in place
__global__ void k_prep(float* __restrict__ agg, const float* __restrict__ x,
                       const float* __restrict__ eps, int layer, int nNodes) {
    int gid = blockIdx.x * blockDim.x + threadIdx.x;
    int node = gid >> 5;
    int lane = gid & 31;
    if (node >= nNodes) return;
    float4 a  = reinterpret_cast<const float4*>(agg + (size_t)node * D)[lane];
    float4 xv = reinterpret_cast<const float4*>(x   + (size_t)node * D)[lane];
    float sa = a.x * a.x + a.y * a.y + a.z * a.z + a.w * a.w;
    float sx = xv.x * xv.x + xv.y * xv.y + xv.z * xv.z + xv.w * xv.w;
    #pragma unroll
    for (int off = 16; off > 0; off >>= 1) {   // wave32 reduction
        sa += __shfl_xor(sa, off, 32);
        sx += __shfl_xor(sx, off, 32);
    }
    float ia = 1.f / fmaxf(sqrtf(sa), 1e-12f);
    float ix = (1.f + eps[layer]) / fmaxf(sqrtf(sx), 1e-12f);
    float4 o;
    o.x = a.x * ia + xv.x * ix;
    o.y = a.y * ia + xv.y * ix;
    o.z = a.z * ia + xv.z * ix;
    o.w = a.w * ia + xv.w * ix;
    reinterpret_cast<float4*>(agg + (size_t)node * D)[lane] = o;
}

// ---------------- H = A @ W + b, fused per-column sum / sum^2 for BN stats
// 8 waves/block; each wave: 16 rows x full N=128 (8 tiles of 16x16)
__global__ void __launch_bounds__(256)
k_gemm_bias_stats(const float* __restrict__ A, const float* __restrict__ W,
                  const float* __restrict__ bias, float* __restrict__ H,
                  float* __restrict__ colsum, float* __restrict__ colsq, int nNodes) {
    __shared__ float sW[D * D];  // 64KB, whole 128x128 weight
    stage_weights(sW, W);

    const int lane  = threadIdx.x & 31;
    const int wave  = threadIdx.x >> 5;
    const int mrow  = lane & 15;   // A: M index / B,C: N index
    const int khalf = lane >> 4;   // which half of the wave
    const int m0    = blockIdx.x * 128 + wave * 16;
    const int m     = m0 + mrow;
    const bool mv   = (m < nNodes);
    const float* arow = A + (size_t)(mv ? m : 0) * D;

    v8f acc[8] = {};
    for (int k = 0; k < D; k += 4) {
        __builtin_prefetch(arow + k + 16, 0, 1);
        float2 av = *reinterpret_cast<const float2*>(arow + k + 2 * khalf);
        v2f af;
        af.x = mv ? av.x : 0.f;  // zero OOB rows -> exact BN stats
        af.y = mv ? av.y : 0.f;
        #pragma unroll
        for (int nt = 0; nt < 8; ++nt) {
            const int n = nt * 16 + mrow;
            v2f bf;
            bf.x = sW[(k + khalf) * D + n];
            bf.y = sW[(k + 2 + khalf) * D + n];
            acc[nt] = wmma4(af, bf, acc[nt]);
        }
    }

    #pragma unroll
    for (int nt = 0; nt < 8; ++nt) {
        const int ncol = nt * 16 + mrow;
        const float b = bias[ncol];
        float s = 0.f, sq = 0.f;
        #pragma unroll
        for (int i = 0; i < 8; ++i) {
            const int row = m0 + khalf * 8 + i;   // C layout: VGPR i, lane half
            float v = acc[nt][i] + b;
            if (row < nNodes) {
                H[(size_t)row * D + ncol] = v;
                s += v;
                sq += v * v;
            }
        }
        atomicAdd(&colsum[ncol], s);
        atomicAdd(&colsq[ncol], sq);
    }
}

// ------------- fold BN stats: aS = gamma*invstd, cS = beta - mean*gamma*invstd
__global__ void k_bnfin(const float* __restrict__ colsum, const float* __restrict__ colsq,
                        const float* __restrict__ gamma, const float* __restrict__ beta,
                        float* __restrict__ aS, float* __restrict__ cS, int nNodes) {
    int n = threadIdx.x;
    float invN = 1.f / (float)nNodes;
    float mu  = colsum[n] * invN;
    float var = colsq[n] * invN - mu * mu;
    float inv = rsqrtf(var + BN_EPS);
    float a = gamma[n] * inv;
    aS[n] = a;
    cS[n] = beta[n] - mu * a;
}

// ---------------- Y = relu?( relu(bn(H)) @ W + b ); BN+ReLU fused into A load
__global__ void __launch_bounds__(256)
k_gemm_bn_relu(const float* __restrict__ Hm, const float* __restrict__ W,
               const float* __restrict__ bias, const float* __restrict__ aS,
               const float* __restrict__ cS, float* __restrict__ Y,
               int nNodes, int reluOut) {
    __shared__ float sW[D * D];
    stage_weights(sW, W);

    const int lane  = threadIdx.x & 31;
    const int wave  = threadIdx.x >> 5;
    const int mrow  = lane & 15;
    const int khalf = lane >> 4;
    const int m0    = blockIdx.x * 128 + wave * 16;
    const int m     = m0 + mrow;
    const bool mv   = (m < nNodes);
    const float* hrow = Hm + (size_t)(mv ? m : 0) * D;

    v8f acc[8] = {};
    for (int k = 0; k < D; k += 4) {
        const int k0 = k + 2 * khalf;
        float2 hv = *reinterpret_cast<const float2*>(hrow + k0);
        v2f af;  // BN + ReLU applied to A operand (garbage rows never stored)
        af.x = fmaxf(fmaf(hv.x, aS[k0],     cS[k0]),     0.f);
        af.y = fmaxf(fmaf(hv.y, aS[k0 + 1], cS[k0 + 1]), 0.f);
        #pragma unroll
        for (int nt = 0; nt < 8; ++nt) {
            const int n = nt * 16 + mrow;
            v2f bf;
            bf.x = sW[(k + khalf) * D + n];
            bf.y = sW[(k + 2 + khalf) * D + n];
            acc[nt] = wmma4(af, bf, acc[nt]);
        }
    }

    #pragma unroll
    for (int nt = 0; nt < 8; ++nt) {
        const int ncol = nt * 16 + mrow;
        const float b = bias[ncol];
        #pragma unroll
        for (int i = 0; i < 8; ++i) {
            const int row = m0 + khalf * 8 + i;
            float v = acc[nt][i] + b;
            if (reluOut) v = fmaxf(v, 0.f);
            if (row < nNodes) Y[(size_t)row * D + ncol] = v;
        }
    }
}

extern "C" void kernel_launch(void* const* d_in, const int* in_sizes, int n_in,
                              void* d_out, int out_size, void* d_ws, size_t ws_size,
                              hipStream_t stream) {
    const float* x0  = (const float*)d_in[0];
    const int*   ei  = (const int*)d_in[1];
    const float* W1  = (const float*)d_in[2];
    const float* b1  = (const float*)d_in[3];
    const float* gm  = (const float*)d_in[4];
    const float* bt  = (const float*)d_in[5];
    const float* W2  = (const float*)d_in[6];
    const float* b2  = (const float*)d_in[7];
    const float* eps = (const float*)d_in[8];

    const int NN = in_sizes[0] / D;   // 50000
    const int E  = in_sizes[1] / 2;   // 800000
    const int* src = ei;
    const int* dst = ei + E;

    float* bufA = (float*)d_ws;                      // agg/out (L0,L1), x (L2)
    float* bufB = bufA + (size_t)NN * D;             // h
    float* bufC = bufB + (size_t)NN * D;             // x1, then agg (L2)
    float* stats  = bufC + (size_t)NN * D;
    float* colsum = stats;
    float* colsq  = stats + D;
    float* aS     = stats + 2 * D;
    float* cS     = stats + 3 * D;

    const int tpb      = 256;
    const int gScatter = (E * 32 + tpb - 1) / tpb;
    const int gPrep    = (NN * 32 + tpb - 1) / tpb;
    const int gGemm    = (NN + 127) / 128;

    const float* xs[3]  = {x0,   bufC, bufA};
    float*       aggs[3]= {bufA, bufA, bufC};
    float*       xns[3] = {bufC, bufA, (float*)d_out};
    const int    relu[3]= {1, 1, 0};

    for (int l = 0; l < 3; ++l) {
        const float* x  = xs[l];
        float* agg      = aggs[l];
        float* xn       = xns[l];

        k_zero<<<1024, tpb, 0, stream>>>(agg, NN * D);
        k_zero<<<1, tpb, 0, stream>>>(colsum, 2 * D);
        k_scatter<<<gScatter, tpb, 0, stream>>>(x, src, dst, agg, E);
        k_prep<<<gPrep, tpb, 0, stream>>>(agg, x, eps, l, NN);
        k_gemm_bias_stats<<<gGemm, tpb, 0, stream>>>(agg, W1 + (size_t)l * D * D,
                                                     b1 + l * D, bufB, colsum, colsq, NN);
        k_bnfin<<<1, D, 0, stream>>>(colsum, colsq, gm + l * D, bt + l * D, aS, cS, NN);
        k_gemm_bn_relu<<<gGemm, tpb, 0, stream>>>(bufB, W2 + (size_t)l * D * D,
                                                  b2 + l * D, aS, cS, xn, NN, relu[l]);
    }
}